// MMD_loss_506806141388
// MI455X (gfx1250) — compile-verified
//
#include <hip/hip_runtime.h>
#include <hip/hip_bf16.h>

typedef float v2f __attribute__((ext_vector_type(2)));
typedef float v8f __attribute__((ext_vector_type(8)));

#define NROWS   8192
#define NHALF   4096
#define DDIM    128
#define TILE    128
#define NTILES  64          // 8192 / 128
#define LDSPAD  132         // 128 + 4 -> conflict-free b64 column-group reads
#define LOG2E   1.4426950408889634f

// ws float layout
#define WS_SCALE    1                     // scale16 = -log2e/(16*bw)
#define WS_COLSUM   16                    // 128 floats
#define WS_SQ       160                   // 8192 floats
#define WS_PARTIAL  (160 + NROWS)         // 4096 floats

// ---------- Phase 1a: per-row squared norms (one wave per row) ----------
__global__ void mmd_rowsq(const float* __restrict__ src,
                          const float* __restrict__ tgt,
                          float* __restrict__ ws) {
    int wave = (blockIdx.x << 3) + (threadIdx.x >> 5);   // 1024 blocks * 8 waves
    int lane = threadIdx.x & 31;
    const float* rowp = (wave < NHALF) ? (src + (size_t)wave * DDIM)
                                       : (tgt + (size_t)(wave - NHALF) * DDIM);
    float4 x = *(const float4*)(rowp + lane * 4);
    float s = x.x * x.x + x.y * x.y + x.z * x.z + x.w * x.w;
    #pragma unroll
    for (int off = 16; off > 0; off >>= 1) s += __shfl_down(s, off);
    if (lane == 0) ws[WS_SQ + wave] = s;
}

// ---------- Phase 1b: column sums (one block per column) ----------
__global__ void mmd_colsum(const float* __restrict__ src,
                           const float* __restrict__ tgt,
                           float* __restrict__ ws) {
    __shared__ float red[256];
    int c = blockIdx.x;
    float s = 0.f;
    for (int r = threadIdx.x; r < NROWS; r += 256) {
        s += (r < NHALF) ? src[(size_t)r * DDIM + c]
                         : tgt[(size_t)(r - NHALF) * DDIM + c];
    }
    red[threadIdx.x] = s;
    __syncthreads();
    for (int h = 128; h > 0; h >>= 1) {
        if (threadIdx.x < h) red[threadIdx.x] += red[threadIdx.x + h];
        __syncthreads();
    }
    if (threadIdx.x == 0) ws[WS_COLSUM + c] = red[0];
}

// ---------- Phase 1c: bandwidth scalar ----------
// sum(L2) = 2*Ns*sum(sq) - 2*||colsum||^2 ; bw = sum(L2)/(Ns^2-Ns)/4
__global__ void mmd_scalars(float* __restrict__ ws) {
    __shared__ float red[256];
    int t = threadIdx.x;
    float s = 0.f;
    for (int i = t; i < NROWS; i += 256) s += ws[WS_SQ + i];
    red[t] = s;
    __syncthreads();
    for (int h = 128; h > 0; h >>= 1) {
        if (t < h) red[t] += red[t + h];
        __syncthreads();
    }
    float sum_sq = red[0];
    __syncthreads();
    float c = (t < DDIM) ? ws[WS_COLSUM + t] : 0.f;
    red[t] = c * c;
    __syncthreads();
    for (int h = 128; h > 0; h >>= 1) {
        if (t < h) red[t] += red[t + h];
        __syncthreads();
    }
    if (t == 0) {
        double s2 = (double)red[0];
        double sumL2 = 2.0 * (double)NROWS * (double)sum_sq - 2.0 * s2;
        double bw = sumL2 / ((double)NROWS * (double)NROWS - (double)NROWS);
        bw *= 0.25;                                  // / KERNEL_MUL^(KERNEL_NUM//2)
        ws[WS_SCALE] = (float)(-(double)LOG2E / (bw * 16.0));
    }
}

// ---------- Phase 2: tiled Gram + multi-scale RBF reduction ----------
__global__ void __launch_bounds__(256)
mmd_tiles(const float* __restrict__ src,
          const float* __restrict__ tgt,
          float* __restrict__ ws) {
    // Single flat LDS block, manually partitioned so the B tile (the one
    // feeding the per-WMMA ds_load_b64 stream) sits at byte offset 0:
    // its addresses are then row_base(VGPR) + 16*kk, and 16*kk <= 496 fits
    // the 16-bit DS immediate -> no per-load address ALU.
    __shared__ float lds[2 * TILE * LDSPAD];
    float (*Bs)[LDSPAD] = (float (*)[LDSPAD])(lds);                  // offset 0
    float (*As)[LDSPAD] = (float (*)[LDSPAD])(lds + TILE * LDSPAD);  // offset 0x10800
    __shared__ float sqA[TILE];
    __shared__ float sqB[TILE];
    __shared__ float red[256];

    int bj = blockIdx.x, bi = blockIdx.y;
    int tid = threadIdx.x;
    float* partial = ws + WS_PARTIAL + bi * NTILES + bj;

    if (bj < bi) {                      // symmetric: only upper triangle of tiles
        if (tid == 0) *partial = 0.f;
        return;
    }

    const float scale16 = ws[WS_SCALE];
    const float* abase = (bi < NTILES / 2) ? (src + (size_t)bi * TILE * DDIM)
                                           : (tgt + (size_t)(bi - NTILES / 2) * TILE * DDIM);
    const float* bbase = (bj < NTILES / 2) ? (src + (size_t)bj * TILE * DDIM)
                                           : (tgt + (size_t)(bj - NTILES / 2) * TILE * DDIM);

    // stage 128x128 B and A tiles (float4 per thread, 16 iterations each)
    for (int p = tid; p < TILE * (DDIM / 4); p += 256) {
        int r = p >> 5;
        int c = (p & 31) << 2;
        *(float4*)&Bs[r][c] = *(const float4*)(bbase + (size_t)r * DDIM + c);
        *(float4*)&As[r][c] = *(const float4*)(abase + (size_t)r * DDIM + c);
    }
    if (tid < TILE) {
        sqA[tid] = ws[WS_SQ + bi * TILE + tid];
        sqB[tid] = ws[WS_SQ + bj * TILE + tid];
    }
    __syncthreads();

    const int wave = tid >> 5;
    const int lane = tid & 31;
    const int half = lane >> 4;         // K-half selector for A/B fragments
    const int lrow = lane & 15;         // M (for A) / N (for B) within subtile
    const int mi = wave * 16;           // this wave's 16-row strip
    const bool diag = (bi == bj);

    float tacc = 0.f;
    const float* arow = &As[mi + lrow][2 * half];

    // Diagonal tiles: subtiles with nj+15 < mi are entirely weight-0; skip
    // them (wave-uniform bound -> EXEC stays all-ones for WMMA).
    const int njt0 = diag ? wave : 0;

    for (int njt = njt0; njt < 8; ++njt) {
        int nj = njt * 16;
        const float* brow = &Bs[nj + lrow][2 * half];
        v8f acc = {0.f, 0.f, 0.f, 0.f, 0.f, 0.f, 0.f, 0.f};
        #pragma unroll
        for (int kk = 0; kk < DDIM / 4; ++kk) {
            v2f a = *(const v2f*)(arow + 4 * kk);    // A[M=lrow][2*half + {0,1}]
            v2f b = *(const v2f*)(brow + 4 * kk);    // B row layout mirrors A
            acc = __builtin_amdgcn_wmma_f32_16x16x4_f32(
                false, a, false, b, (short)0, acc, false, false);
        }
        // epilogue: D layout -> VGPR v holds (M = v + 8*half, N = lrow)
        #pragma unroll
        for (int v = 0; v < 8; ++v) {
            int gi = mi + v + 8 * half;
            int gj = nj + lrow;
            float L2 = sqA[gi] + sqB[gj] - 2.0f * acc[v];
            float wgt = diag ? ((gj > gi) ? 2.0f : ((gj == gi) ? 1.0f : 0.0f))
                             : 2.0f;
            // sum_{i=0..4} exp(-L2/(bw*2^i)) = u + u^2 + u^4 + u^8 + u^16,
            // u = exp2(-L2*log2e/(16*bw))
            float u   = __builtin_amdgcn_exp2f(L2 * scale16);
            float p2  = u * u;
            float p4  = p2 * p2;
            float p8  = p4 * p4;
            float p16 = p8 * p8;
            tacc += wgt * (u + p2 + p4 + p8 + p16);
        }
    }

    float sign = ((bi < NTILES / 2) == (bj < NTILES / 2)) ? 1.0f : -1.0f;
    red[tid] = tacc * sign;
    __syncthreads();
    for (int h = 128; h > 0; h >>= 1) {
        if (tid < h) red[tid] += red[tid + h];
        __syncthreads();
    }
    if (tid == 0) *partial = red[0];
}

// ---------- Phase 3: deterministic final reduction ----------
__global__ void mmd_final(const float* __restrict__ ws, float* __restrict__ out) {
    __shared__ float red[256];
    int t = threadIdx.x;
    float s = 0.f;
    for (int i = t; i < NTILES * NTILES; i += 256) s += ws[WS_PARTIAL + i];
    red[t] = s;
    __syncthreads();
    for (int h = 128; h > 0; h >>= 1) {
        if (t < h) red[t] += red[t + h];
        __syncthreads();
    }
    if (t == 0) out[0] = red[0] * (1.0f / ((float)NHALF * (float)NHALF));
}

extern "C" void kernel_launch(void* const* d_in, const int* in_sizes, int n_in,
                              void* d_out, int out_size, void* d_ws, size_t ws_size,
                              hipStream_t stream) {
    const float* src = (const float*)d_in[0];
    const float* tgt = (const float*)d_in[1];
    float* out = (float*)d_out;
    float* ws  = (float*)d_ws;

    mmd_rowsq  <<<NROWS / 8, 256, 0, stream>>>(src, tgt, ws);
    mmd_colsum <<<DDIM,      256, 0, stream>>>(src, tgt, ws);
    mmd_scalars<<<1,         256, 0, stream>>>(ws);
    mmd_tiles  <<<dim3(NTILES, NTILES), 256, 0, stream>>>(src, tgt, ws);
    mmd_final  <<<1,         256, 0, stream>>>(ws, out);
}